// DPCN_28767690948915
// MI455X (gfx1250) — compile-verified
//
#include <hip/hip_runtime.h>
#include <hip/hip_bf16.h>
#include <cstdint>

// ---------------- problem constants ----------------
constexpr int B = 4, N = 8192, S = 1024, K = 16;
constexpr int SK = S * K;               // columns per batch in pcconv space
constexpr int CPAD = 544;               // 513 padded to multiple of 32 (K-stage depth)

#define CDIV(a, b) (((a) + (b) - 1) / (b))

typedef __attribute__((ext_vector_type(2))) float v2f;
typedef __attribute__((ext_vector_type(8))) float v8f;

// ================= FPS: one block per batch =================
__global__ void k_fps(const float* __restrict__ xyz, int* __restrict__ idx) {
  __shared__ float dmin[N];
  __shared__ float rb[256];
  __shared__ int   ri[256];
  __shared__ int   s_last;
  const int b = blockIdx.x, t = threadIdx.x;
  const float* px = xyz + (size_t)b * 3 * N;
  for (int i = t; i < N; i += 256) dmin[i] = 1e10f;
  if (t == 0) { s_last = 0; idx[b * S] = 0; }
  __syncthreads();
  for (int it = 1; it < S; ++it) {
    const int last = s_last;
    const float lx = px[last], ly = px[N + last], lz = px[2 * N + last];
    float best = -1.0f; int bi = 0x7fffffff;
    for (int i = t; i < N; i += 256) {
      float dx = px[i] - lx, dy = px[N + i] - ly, dz = px[2 * N + i] - lz;
      float d = dx * dx + dy * dy + dz * dz;
      float dm = dmin[i];
      if (d < dm) { dm = d; dmin[i] = d; }
      if (dm > best) { best = dm; bi = i; }
    }
    rb[t] = best; ri[t] = bi;
    __syncthreads();
    for (int off = 128; off > 0; off >>= 1) {
      if (t < off) {
        if (rb[t + off] > rb[t] || (rb[t + off] == rb[t] && ri[t + off] < ri[t])) {
          rb[t] = rb[t + off]; ri[t] = ri[t + off];
        }
      }
      __syncthreads();
    }
    if (t == 0) { s_last = ri[0]; idx[b * S + it] = ri[0]; }
    __syncthreads();
  }
}

// gather sampled centers: cxyz[b,3,S]
__global__ void k_centers(const float* __restrict__ xyz, const int* __restrict__ idx,
                          float* __restrict__ cx) {
  int g = blockIdx.x * blockDim.x + threadIdx.x;
  if (g >= B * 3 * S) return;
  int s = g % S, d = (g / S) % 3, b = g / (3 * S);
  cx[(b * 3 + d) * S + s] = xyz[(size_t)(b * 3 + d) * N + idx[b * S + s]];
}

// ================= KNN (K=16): 8 centers per 256-thread block, 1 wave each ===
__global__ void k_knn(const float* __restrict__ xyz, const float* __restrict__ cx,
                      int* __restrict__ ind) {
  __shared__ float cd[8][512];
  __shared__ int   ci[8][512];
  const int t = threadIdx.x, w = t >> 5, l = t & 31;
  const int sg = blockIdx.x * 8 + w;
  if (sg < B * S) {
    const int b = sg / S, s = sg % S;
    const float* px = xyz + (size_t)b * 3 * N;
    const float cxv = cx[(b * 3 + 0) * S + s];
    const float cyv = cx[(b * 3 + 1) * S + s];
    const float czv = cx[(b * 3 + 2) * S + s];
    float* md = cd[w] + l * 16;
    int*   mi = ci[w] + l * 16;
    for (int j = 0; j < 16; ++j) { md[j] = 3.4e38f; mi[j] = 0x7fffffff; }
    float curmax = 3.4e38f; int curslot = 0;
    for (int i = l; i < N; i += 32) {
      float dx = px[i] - cxv, dy = px[N + i] - cyv, dz = px[2 * N + i] - czv;
      float d = dx * dx + dy * dy + dz * dz;
      if (d < curmax) {
        md[curslot] = d; mi[curslot] = i;
        curmax = -1.0f;
        for (int j = 0; j < 16; ++j) if (md[j] > curmax) { curmax = md[j]; curslot = j; }
      }
    }
  }
  __syncthreads();
  if (sg < B * S && l == 0) {
    const int b = sg / S, s = sg % S;
    for (int k = 0; k < K; ++k) {
      float bd = 3.4e38f; int bi = 0x7fffffff, bp = 0;
      for (int j = 0; j < 512; ++j) {
        float d = cd[w][j]; int id = ci[w][j];
        if (d < bd || (d == bd && id < bi)) { bd = d; bi = id; bp = j; }
      }
      ind[(size_t)(b * S + s) * K + k] = bi;
      cd[w][bp] = 3.4e38f; ci[w][bp] = 0x7fffffff;
    }
  }
}

// grouped relative coordinates g[b,3,S,K]
__global__ void k_group_xyz(const float* __restrict__ xyz, const float* __restrict__ cx,
                            const int* __restrict__ ind, float* __restrict__ g) {
  int gi = blockIdx.x * blockDim.x + threadIdx.x;
  if (gi >= B * S * K) return;
  int k = gi % K, s = (gi / K) % S, b = gi / (S * K);
  int i = ind[(size_t)(b * S + s) * K + k];
  for (int d = 0; d < 3; ++d)
    g[((size_t)(b * 3 + d) * S + s) * K + k] =
        xyz[(size_t)(b * 3 + d) * N + i] - cx[(b * 3 + d) * S + s];
}

// Fsum[b,s,k] = sum_c feat[b,c,clamp(ind)]   (JAX clamps OOB gathers)
__global__ void k_fsum(const float* __restrict__ feat, int featN, int C,
                       const int* __restrict__ ind, float* __restrict__ fsum) {
  int gi = blockIdx.x * blockDim.x + threadIdx.x;
  if (gi >= B * S * K) return;
  int k = gi % K, s = (gi / K) % S, b = gi / (S * K);
  int i = ind[(size_t)(b * S + s) * K + k];
  if (i > featN - 1) i = featN - 1;
  float a = 0.f;
  for (int c = 0; c < C; ++c) a += feat[(size_t)(b * C + c) * featN + i];
  fsum[(size_t)(b * S + s) * K + k] = a;
}

// small channel-mixing conv over grouped columns: Y[b,O,cols] = W[O,Cin] X[b,Cin,cols] + b
__global__ void k_conv_cols(float* __restrict__ Y, const float* __restrict__ X,
                            const float* __restrict__ Wm, const float* __restrict__ bias,
                            int O, int Cin, int cols) {
  int gi = blockIdx.x * blockDim.x + threadIdx.x;
  if (gi >= B * cols) return;
  int cc = gi % cols, b = gi / cols;
  for (int o = 0; o < O; ++o) {
    float acc = bias[o];
    for (int ci = 0; ci < Cin; ++ci)
      acc += Wm[o * Cin + ci] * X[((size_t)b * Cin + ci) * cols + cc];
    Y[((size_t)b * O + o) * cols + cc] = acc;
  }
}

// ================= train-mode BatchNorm (per channel over B*M) =================
__global__ void k_bn_stats(const float* __restrict__ T, int C, int M,
                           float* __restrict__ sum, float* __restrict__ sumsq) {
  __shared__ float rs[256], rq[256];
  const int bc = blockIdx.x, c = bc % C, t = threadIdx.x;
  const float* p = T + (size_t)bc * M;
  float s = 0.f, q = 0.f;
  for (int i = t; i < M; i += 256) { float v = p[i]; s += v; q += v * v; }
  rs[t] = s; rq[t] = q;
  __syncthreads();
  for (int off = 128; off > 0; off >>= 1) {
    if (t < off) { rs[t] += rs[t + off]; rq[t] += rq[t + off]; }
    __syncthreads();
  }
  if (t == 0) { atomicAdd(&sum[c], rs[0]); atomicAdd(&sumsq[c], rq[0]); }
}

__global__ void k_bn_finalize(const float* __restrict__ sum, const float* __restrict__ sumsq,
                              float* __restrict__ mean, float* __restrict__ inv,
                              int C, float cnt) {
  int c = blockIdx.x * blockDim.x + threadIdx.x;
  if (c >= C) return;
  float m = sum[c] / cnt;
  float v = sumsq[c] / cnt - m * m;
  mean[c] = m;
  inv[c] = rsqrtf(v + 1e-3f);
}

__global__ void k_bn_apply(float* __restrict__ T, int C, int M,
                           const float* __restrict__ mean, const float* __restrict__ inv,
                           const float* __restrict__ gamma, const float* __restrict__ beta,
                           int relu, size_t total) {
  size_t gi = (size_t)blockIdx.x * blockDim.x + threadIdx.x;
  if (gi >= total) return;
  int c = (int)((gi / M) % C);
  float v = (T[gi] - mean[c]) * inv[c] * gamma[c] + beta[c];
  T[gi] = (relu && v < 0.f) ? 0.f : v;
}

// message: f[b,o,s] (+)= (1/K) sum_k Wt[b,o,s,k] * Fsum[b,s,k]
__global__ void k_message(const float* __restrict__ Wt, const float* __restrict__ fsum,
                          float* __restrict__ f, int O, int accumulate) {
  int gi = blockIdx.x * blockDim.x + threadIdx.x;
  if (gi >= B * S * 128 || gi >= B * S * O) return;
  int o = gi % O, s = (gi / O) % S, b = gi / (O * S);
  float acc = 0.f;
  const float* wp = Wt + ((size_t)(b * O + o) * S + s) * K;
  const float* fp = fsum + ((size_t)b * S + s) * K;
  #pragma unroll
  for (int k = 0; k < K; ++k) acc += wp[k] * fp[k];
  acc *= (1.0f / K);
  size_t oi = ((size_t)b * O + o) * S + s;
  if (accumulate) f[oi] += acc; else f[oi] = acc;
}

__global__ void k_maxpool(const float* __restrict__ f, float* __restrict__ pool) {
  int gi = blockIdx.x * blockDim.x + threadIdx.x;
  if (gi >= B * 128) return;
  const float* p = f + (size_t)gi * S;
  float m = p[0];
  for (int s = 1; s < S; ++s) m = fmaxf(m, p[s]);
  pool[gi] = m;
}

// tiny FC + train-mode BN over batch dim (B=4) + ReLU, one thread per channel
__global__ void k_fc_bn(const float* __restrict__ x, const float* __restrict__ Wm,
                        const float* __restrict__ bias, const float* __restrict__ gamma,
                        const float* __restrict__ beta, float* __restrict__ y,
                        int Cin, int O) {
  int o = blockIdx.x * blockDim.x + threadIdx.x;
  if (o >= O) return;
  float pre[B];
  for (int b = 0; b < B; ++b) {
    float acc = bias[o];
    for (int c = 0; c < Cin; ++c) acc += Wm[o * Cin + c] * x[b * Cin + c];
    pre[b] = acc;
  }
  float m = 0.f, q = 0.f;
  for (int b = 0; b < B; ++b) { m += pre[b]; q += pre[b] * pre[b]; }
  m *= (1.0f / B);
  float inv = rsqrtf(q / B - m * m + 1e-3f);
  for (int b = 0; b < B; ++b) {
    float v = (pre[b] - m) * inv * gamma[o] + beta[o];
    y[b * O + o] = v < 0.f ? 0.f : v;
  }
}

// feat_final[b,512,S] = concat(p1, f1, p2, f2)
__global__ void k_feat_final(const float* __restrict__ p1, const float* __restrict__ f1,
                             const float* __restrict__ p2, const float* __restrict__ f2,
                             float* __restrict__ ff) {
  int gi = blockIdx.x * blockDim.x + threadIdx.x;
  if (gi >= B * 512 * S) return;
  int s = gi % S, ch = (gi / S) % 512, b = gi / (512 * S);
  float v;
  if (ch < 128)      v = p1[b * 128 + ch];
  else if (ch < 256) v = f1[((size_t)b * 128 + (ch - 128)) * S + s];
  else if (ch < 384) v = p2[b * 128 + (ch - 256)];
  else               v = f2[((size_t)b * 128 + (ch - 384)) * S + s];
  ff[(size_t)gi] = v;
}

// 3-NN of each point against S centers + inverse-distance weights
__global__ void k_fp_knn3(const float* __restrict__ xyz, const float* __restrict__ cx,
                          int* __restrict__ idx3, float* __restrict__ w3) {
  __shared__ float cxl[S], cyl[S], czl[S];
  const int b = blockIdx.y, t = threadIdx.x;
  for (int i = t; i < S; i += 256) {
    cxl[i] = cx[(b * 3 + 0) * S + i];
    cyl[i] = cx[(b * 3 + 1) * S + i];
    czl[i] = cx[(b * 3 + 2) * S + i];
  }
  __syncthreads();
  const int n = blockIdx.x * 256 + t;
  const float px = xyz[(size_t)(b * 3 + 0) * N + n];
  const float py = xyz[(size_t)(b * 3 + 1) * N + n];
  const float pz = xyz[(size_t)(b * 3 + 2) * N + n];
  float d0 = 3.4e38f, d1 = 3.4e38f, d2 = 3.4e38f;
  int i0 = 0, i1 = 0, i2 = 0;
  for (int s = 0; s < S; ++s) {
    float dx = px - cxl[s], dy = py - cyl[s], dz = pz - czl[s];
    float d = dx * dx + dy * dy + dz * dz;
    if (d < d0)      { d2 = d1; i2 = i1; d1 = d0; i1 = i0; d0 = d; i0 = s; }
    else if (d < d1) { d2 = d1; i2 = i1; d1 = d; i1 = s; }
    else if (d < d2) { d2 = d; i2 = s; }
  }
  float w0 = 1.0f / (d0 + 1e-10f), w1 = 1.0f / (d1 + 1e-10f), w2 = 1.0f / (d2 + 1e-10f);
  float ws = w0 + w1 + w2;
  size_t base = ((size_t)b * N + n) * 3;
  idx3[base + 0] = i0; idx3[base + 1] = i1; idx3[base + 2] = i2;
  w3[base + 0] = w0 / ws; w3[base + 1] = w1 / ws; w3[base + 2] = w2 / ws;
}

// interpolate 512 channels into padded X buffer [B, CPAD, N]
__global__ void k_interp(const float* __restrict__ ff, const int* __restrict__ idx3,
                         const float* __restrict__ w3, float* __restrict__ X) {
  size_t gi = (size_t)blockIdx.x * blockDim.x + threadIdx.x;
  if (gi >= (size_t)B * 512 * N) return;
  int n = (int)(gi % N); int ch = (int)((gi / N) % 512); int b = (int)(gi / ((size_t)512 * N));
  size_t base = ((size_t)b * N + n) * 3;
  const float* fc = ff + ((size_t)b * 512 + ch) * S;
  float acc = fc[idx3[base]] * w3[base] + fc[idx3[base + 1]] * w3[base + 1] +
              fc[idx3[base + 2]] * w3[base + 2];
  X[((size_t)b * CPAD + ch) * N + n] = acc;
}

// channel 512 = feat1, channels 513..CPAD-1 = 0
__global__ void k_xbuf_tail(const float* __restrict__ feat1, float* __restrict__ X) {
  int gi = blockIdx.x * blockDim.x + threadIdx.x;
  if (gi >= B * N) return;
  int n = gi % N, b = gi / N;
  X[((size_t)b * CPAD + 512) * N + n] = feat1[(size_t)b * N + n];
  for (int c = 513; c < CPAD; ++c)
    X[((size_t)b * CPAD + c) * N + n] = 0.f;
}

__global__ void k_wpad(const float* __restrict__ w, float* __restrict__ wp) {
  int gi = blockIdx.x * blockDim.x + threadIdx.x;
  if (gi >= 256 * CPAD) return;
  int c = gi % CPAD, o = gi / CPAD;
  wp[gi] = (c < 513) ? w[o * 513 + c] : 0.f;
}

// ============ f32 WMMA GEMM with async-to-LDS double-buffered X staging ============
// Block = 128 threads = 4 waves. Block covers a 64(O) x 16(N) output panel:
// wave w handles rows ot = blockIdx.y*64 + w*16. The 32(K) x 16(N) X slice is
// staged once per block into LDS via GLOBAL_LOAD_ASYNC_TO_LDS_B32 (ASYNCcnt),
// double-buffered, and consumed by all 4 waves (4x reuse). A-fragments stream
// from global with global_prefetch ahead of use. Cin must be a multiple of 32.
__global__ void k_gemm_wmma(const float* __restrict__ Wm, const float* __restrict__ X,
                            const float* __restrict__ bias, float* __restrict__ Y,
                            int Cin, int lda, int Ncols, int xbstride, int ybstride) {
  __shared__ float xs[2][32 * 16];      // 4 KB double buffer
  const int t  = threadIdx.x;           // 0..127
  const int w  = t >> 5;                // wave in block
  const int l  = t & 31;                // lane
  const int m  = l & 15;                // A-row / B-col / D-col index for this lane
  const int hf = l >> 4;                // half selects K pair (A/B) and M half (C/D)
  const int nt = blockIdx.x * 16;
  const int ot = blockIdx.y * 64 + w * 16;
  const int b  = blockIdx.z;
  const float* Xb = X + (size_t)b * xbstride;
  float* Yb = Y + (size_t)b * ybstride;

  // stage one 32x16 K-slice (512 floats) into xs[buf], 4 async b32 loads/thread
  auto stage = [&](int buf, int kk) {
    #pragma unroll
    for (int i = 0; i < 4; ++i) {
      int e  = t + i * 128;
      int kr = e >> 4, nc = e & 15;
      const float* gp = Xb + (size_t)(kk + kr) * Ncols + nt + nc;
      uint32_t la = (uint32_t)(uintptr_t)&xs[buf][e];   // LDS byte address (low 32 bits of pointer)
      asm volatile("global_load_async_to_lds_b32 %0, %1, off"
                   :: "v"(la), "v"((uint64_t)(uintptr_t)gp) : "memory");
    }
  };

  v8f acc;
  #pragma unroll
  for (int r = 0; r < 8; ++r) acc[r] = bias[ot + r + 8 * hf];

  stage(0, 0);
  int cur = 0;
  for (int kk = 0; kk < Cin; kk += 32) {
    const bool has_next = (kk + 32) < Cin;
    if (has_next) {
      stage(cur ^ 1, kk + 32);
      asm volatile("s_wait_asynccnt 4" ::: "memory");   // drain current slice
    } else {
      asm volatile("s_wait_asynccnt 0" ::: "memory");
    }
    __syncthreads();
    __builtin_prefetch(Wm + (size_t)(ot + m) * lda + kk + 32, 0, 0);
    #pragma unroll
    for (int ks = 0; ks < 8; ++ks) {
      const int ka = ks * 4 + 2 * hf;
      v2f av, bv;
      av[0] = Wm[(size_t)(ot + m) * lda + kk + ka];
      av[1] = Wm[(size_t)(ot + m) * lda + kk + ka + 1];
      bv[0] = xs[cur][ka * 16 + m];
      bv[1] = xs[cur][(ka + 1) * 16 + m];
      acc = __builtin_amdgcn_wmma_f32_16x16x4_f32(false, av, false, bv,
                                                  (short)0, acc, false, false);
    }
    __syncthreads();   // slice consumed by all waves before next overwrite
    cur ^= 1;
  }
  #pragma unroll
  for (int r = 0; r < 8; ++r)
    Yb[(size_t)(ot + r + 8 * hf) * Ncols + nt + m] = acc[r];
}

// final 128 -> 3 conv (no BN, no relu)
__global__ void k_conv3(const float* __restrict__ h, const float* __restrict__ w,
                        const float* __restrict__ bias, float* __restrict__ out) {
  int gi = blockIdx.x * blockDim.x + threadIdx.x;
  if (gi >= B * N) return;
  int n = gi % N, b = gi / N;
  for (int o = 0; o < 3; ++o) {
    float acc = bias[o];
    for (int c = 0; c < 128; ++c)
      acc += w[o * 128 + c] * h[((size_t)b * 128 + c) * N + n];
    out[((size_t)b * 3 + o) * N + n] = acc;
  }
}

// ================= host-side orchestration =================
static void run_bn(float* T, int C, int Mpb, const float* gamma, const float* beta,
                   float* stats, int relu, hipStream_t stream) {
  hipMemsetAsync(stats, 0, 2 * 1024 * sizeof(float), stream);
  k_bn_stats<<<B * C, 256, 0, stream>>>(T, C, Mpb, stats, stats + 1024);
  k_bn_finalize<<<CDIV(C, 256), 256, 0, stream>>>(stats, stats + 1024, stats + 2048,
                                                  stats + 3072, C, (float)B * (float)Mpb);
  size_t tot = (size_t)B * C * Mpb;
  k_bn_apply<<<(unsigned)CDIV(tot, 256), 256, 0, stream>>>(T, C, Mpb, stats + 2048,
                                                           stats + 3072, gamma, beta, relu, tot);
}

extern "C" void kernel_launch(void* const* d_in, const int* in_sizes, int n_in,
                              void* d_out, int out_size, void* d_ws, size_t ws_size,
                              hipStream_t stream) {
  // -------- inputs (setup_inputs dict order; params in insertion order) --------
  const float* xyz1  = (const float*)d_in[0];
  const float* xyz2  = (const float*)d_in[1];
  const float* feat1 = (const float*)d_in[2];
  const float* feat2 = (const float*)d_in[3];
  int pi = 4;
  const float* pc_w1 = (const float*)d_in[pi++]; const float* pc_b1 = (const float*)d_in[pi++];
  const float* pc_g1 = (const float*)d_in[pi++]; const float* pc_e1 = (const float*)d_in[pi++];
  const float* pc_w2 = (const float*)d_in[pi++]; const float* pc_b2 = (const float*)d_in[pi++];
  const float* pc_g2 = (const float*)d_in[pi++]; const float* pc_e2 = (const float*)d_in[pi++];
  const float* pl_w1 = (const float*)d_in[pi++]; const float* pl_b1 = (const float*)d_in[pi++];
  const float* pl_g1 = (const float*)d_in[pi++]; const float* pl_e1 = (const float*)d_in[pi++];
  const float* pl_w2 = (const float*)d_in[pi++]; const float* pl_b2 = (const float*)d_in[pi++];
  const float* pl_g2 = (const float*)d_in[pi++]; const float* pl_e2 = (const float*)d_in[pi++];
  const float* lin_w1 = (const float*)d_in[pi++]; const float* lin_b1 = (const float*)d_in[pi++];
  const float* lin_g1 = (const float*)d_in[pi++]; const float* lin_e1 = (const float*)d_in[pi++];
  const float* lin_w2 = (const float*)d_in[pi++]; const float* lin_b2 = (const float*)d_in[pi++];
  const float* lin_g2 = (const float*)d_in[pi++]; const float* lin_e2 = (const float*)d_in[pi++];
  const float* fp_w1 = (const float*)d_in[pi++]; const float* fp_b1 = (const float*)d_in[pi++];
  const float* fp_g1 = (const float*)d_in[pi++]; const float* fp_e1 = (const float*)d_in[pi++];
  const float* fp_w2 = (const float*)d_in[pi++]; const float* fp_b2 = (const float*)d_in[pi++];
  const float* fp_g2 = (const float*)d_in[pi++]; const float* fp_e2 = (const float*)d_in[pi++];
  const float* cl_w1 = (const float*)d_in[pi++]; const float* cl_b1 = (const float*)d_in[pi++];
  const float* cl_g1 = (const float*)d_in[pi++]; const float* cl_e1 = (const float*)d_in[pi++];
  const float* cl_w2 = (const float*)d_in[pi++]; const float* cl_b2 = (const float*)d_in[pi++];
  const float* cl_g2 = (const float*)d_in[pi++]; const float* cl_e2 = (const float*)d_in[pi++];
  const float* cl_w3 = (const float*)d_in[pi++]; const float* cl_b3 = (const float*)d_in[pi++];

  // -------- workspace layout (4-byte words) --------
  float* W = (float*)d_ws;
  size_t o = 0;
  int*   fps_idx = (int*)(W + o);   o += (size_t)B * S;
  int*   ind1    = (int*)(W + o);   o += (size_t)B * S * K;
  int*   ind2    = (int*)(W + o);   o += (size_t)B * S * K;
  float* cxyz    = W + o;           o += (size_t)B * 3 * S;
  float* g1      = W + o;           o += (size_t)B * 3 * SK;
  float* g2      = W + o;           o += (size_t)B * 3 * SK;
  float* fsum    = W + o;           o += (size_t)B * SK;
  float* preA    = W + o;           o += (size_t)B * 64 * SK;      // up to 64 ch
  float* preB    = W + o;           o += (size_t)B * 128 * SK;     // up to 128 ch
  float* f1      = W + o;           o += (size_t)B * 32 * S;
  float* f2      = W + o;           o += (size_t)B * 32 * S;
  float* f1h     = W + o;           o += (size_t)B * 128 * S;
  float* f2h     = W + o;           o += (size_t)B * 128 * S;
  float* pool    = W + o;           o += (size_t)B * 128;
  float* linh    = W + o;           o += (size_t)B * 256;
  float* p1      = W + o;           o += (size_t)B * 128;
  float* p2      = W + o;           o += (size_t)B * 128;
  float* ff      = W + o;           o += (size_t)B * 512 * S;
  int*   idx3    = (int*)(W + o);   o += (size_t)B * N * 3;
  float* w3      = W + o;           o += (size_t)B * N * 3;
  float* Xbuf    = W + o;           o += (size_t)B * CPAD * N;
  float* Wpad    = W + o;           o += (size_t)256 * CPAD;
  float* h1      = W + o;           o += (size_t)B * 256 * N;
  float* h2      = W + o;           o += (size_t)B * 256 * N;
  float* stats   = W + o;           o += 4 * 1024;                 // sum/sumsq/mean/inv

  // -------- 1. FPS + centers --------
  k_fps<<<B, 256, 0, stream>>>(xyz1, fps_idx);
  k_centers<<<CDIV(B * 3 * S, 256), 256, 0, stream>>>(xyz1, fps_idx, cxyz);

  // -------- 2. KNN + grouping --------
  k_knn<<<CDIV(B * S, 8), 256, 0, stream>>>(xyz1, cxyz, ind1);
  k_knn<<<CDIV(B * S, 8), 256, 0, stream>>>(xyz2, cxyz, ind2);
  k_group_xyz<<<CDIV(B * S * K, 256), 256, 0, stream>>>(xyz1, cxyz, ind1, g1);
  k_group_xyz<<<CDIV(B * S * K, 256), 256, 0, stream>>>(xyz2, cxyz, ind2, g2);

  // -------- 3. seven pcconv layers (weights depend only on g) --------
  const float* gs[2]   = {g1, g2};
  const int*   inds[2] = {ind1, ind2};
  float*       fs[2]   = {f1, f2};
  const float* raw[2]  = {feat1, feat2};
  for (int layer = 0; layer < 7; ++layer) {
    const float* w1 = pc_w1 + layer * 16 * 3;  const float* b1 = pc_b1 + layer * 16;
    const float* ga = pc_g1 + layer * 16;      const float* ea = pc_e1 + layer * 16;
    const float* w2 = pc_w2 + layer * 32 * 16; const float* b2 = pc_b2 + layer * 32;
    const float* gb = pc_g2 + layer * 32;      const float* eb = pc_e2 + layer * 32;
    for (int c = 0; c < 2; ++c) {
      k_conv_cols<<<CDIV(B * SK, 256), 256, 0, stream>>>(preA, gs[c], w1, b1, 16, 3, SK);
      run_bn(preA, 16, SK, ga, ea, stats, 1, stream);
      k_conv_cols<<<CDIV(B * SK, 256), 256, 0, stream>>>(preB, preA, w2, b2, 32, 16, SK);
      run_bn(preB, 32, SK, gb, eb, stats, 1, stream);
      if (layer == 0)
        k_fsum<<<CDIV(B * S * K, 256), 256, 0, stream>>>(raw[c], N, 1, inds[c], fsum);
      else
        k_fsum<<<CDIV(B * S * K, 256), 256, 0, stream>>>(fs[c], S, 32, inds[c], fsum);
      k_message<<<CDIV(B * S * 32, 256), 256, 0, stream>>>(preB, fsum, fs[c], 32, layer > 0);
    }
  }

  // -------- 4. final pcconv (3->64->128 weight MLP) --------
  float* fh[2] = {f1h, f2h};
  for (int c = 0; c < 2; ++c) {
    k_conv_cols<<<CDIV(B * SK, 256), 256, 0, stream>>>(preA, gs[c], pl_w1, pl_b1, 64, 3, SK);
    run_bn(preA, 64, SK, pl_g1, pl_e1, stats, 1, stream);
    k_conv_cols<<<CDIV(B * SK, 256), 256, 0, stream>>>(preB, preA, pl_w2, pl_b2, 128, 64, SK);
    run_bn(preB, 128, SK, pl_g2, pl_e2, stats, 1, stream);
    k_fsum<<<CDIV(B * S * K, 256), 256, 0, stream>>>(fs[c], S, 32, inds[c], fsum);
    k_message<<<CDIV(B * S * 128, 256), 256, 0, stream>>>(preB, fsum, fh[c], 128, 0);
  }

  // -------- 5. global pooling + lin blocks --------
  float* pv[2] = {p1, p2};
  for (int c = 0; c < 2; ++c) {
    k_maxpool<<<CDIV(B * 128, 128), 128, 0, stream>>>(fh[c], pool);
    k_fc_bn<<<1, 256, 0, stream>>>(pool, lin_w1, lin_b1, lin_g1, lin_e1, linh, 128, 256);
    k_fc_bn<<<1, 256, 0, stream>>>(linh, lin_w2, lin_b2, lin_g2, lin_e2, pv[c], 256, 128);
  }
  k_feat_final<<<CDIV(B * 512 * S, 256), 256, 0, stream>>>(p1, f1h, p2, f2h, ff);

  // -------- 6. FeaturePropagation: 3-NN interpolation --------
  k_fp_knn3<<<dim3(N / 256, B), 256, 0, stream>>>(xyz1, cxyz, idx3, w3);
  k_interp<<<(unsigned)CDIV((size_t)B * 512 * N, 256), 256, 0, stream>>>(ff, idx3, w3, Xbuf);
  k_xbuf_tail<<<CDIV(B * N, 256), 256, 0, stream>>>(feat1, Xbuf);
  k_wpad<<<CDIV(256 * CPAD, 256), 256, 0, stream>>>(fp_w1, Wpad);

  // -------- 7. FP / classifier conv chain on the WMMA pipe --------
  // h1 = relu(BN(fp_w1 * [interp; feat1]))
  k_gemm_wmma<<<dim3(N / 16, 256 / 64, B), 128, 0, stream>>>(Wpad, Xbuf, fp_b1, h1,
      CPAD, CPAD, N, CPAD * N, 256 * N);
  run_bn(h1, 256, N, fp_g1, fp_e1, stats, 1, stream);
  // h2 = relu(BN(fp_w2 * h1))
  k_gemm_wmma<<<dim3(N / 16, 256 / 64, B), 128, 0, stream>>>(fp_w2, h1, fp_b2, h2,
      256, 256, N, 256 * N, 256 * N);
  run_bn(h2, 256, N, fp_g2, fp_e2, stats, 1, stream);
  // h1 = relu(BN(cl_w1 * h2))
  k_gemm_wmma<<<dim3(N / 16, 256 / 64, B), 128, 0, stream>>>(cl_w1, h2, cl_b1, h1,
      256, 256, N, 256 * N, 256 * N);
  run_bn(h1, 256, N, cl_g1, cl_e1, stats, 1, stream);
  // h2 = relu(BN(cl_w2 * h1))  (128 channels)
  k_gemm_wmma<<<dim3(N / 16, 128 / 64, B), 128, 0, stream>>>(cl_w2, h1, cl_b2, h2,
      256, 256, N, 256 * N, 128 * N);
  run_bn(h2, 128, N, cl_g2, cl_e2, stats, 1, stream);
  // flow = cl_w3 * h2 + cl_b3
  k_conv3<<<CDIV(B * N, 256), 256, 0, stream>>>(h2, cl_w3, cl_b3, (float*)d_out);

  (void)in_sizes; (void)n_in; (void)out_size; (void)ws_size; (void)o;
}